// KWinnersTakeAll_9328668967564
// MI455X (gfx1250) — compile-verified
//
#include <hip/hip_runtime.h>

#define COLS       4096
#define K_ACTIVE   205        // ceil(0.05 * 4096)
#define THREADS    256
#define PER_THREAD (COLS / THREADS)   // 16

typedef unsigned int uint32x4_t __attribute__((ext_vector_type(4)));
typedef int          int32x4_t  __attribute__((ext_vector_type(4)));
typedef int          int32x8_t  __attribute__((ext_vector_type(8)));

#if defined(__has_builtin)
#if __has_builtin(__builtin_amdgcn_tensor_load_to_lds) && \
    __has_builtin(__builtin_amdgcn_tensor_store_from_lds) && \
    __has_builtin(__builtin_amdgcn_s_wait_tensorcnt)
#define USE_TDM 1
#endif
#endif

__global__ __launch_bounds__(THREADS) void KWinnersTakeAll_kernel(
    const float* __restrict__ x, float* __restrict__ out) {
  // 16 KB row staging + 1 KB histogram + control words (~17.4 KB of WGP's 320 KB)
  __shared__ unsigned s_data[COLS];
  __shared__ unsigned s_hist[256];
  __shared__ unsigned s_prefix;
  __shared__ unsigned s_krem;

  const unsigned t   = threadIdx.x;
  const unsigned row = blockIdx.x;

  if (t == 0) { s_prefix = 0u; s_krem = K_ACTIVE; }

#ifdef USE_TDM
  // ---- Wave 0 issues one TDM descriptor: DMA the whole row (4096 x f32) into LDS.
  if (t < 32) {
    unsigned long long ga  = (unsigned long long)(x + (size_t)row * COLS);
    unsigned lds_off       = (unsigned)(unsigned long long)(s_data);  // low 32b = LDS byte offset

    uint32x4_t g0;
    g0.x = 1u;                                      // count=1 (valid user descriptor)
    g0.y = lds_off;                                 // lds_addr (bytes)
    g0.z = (unsigned)(ga & 0xFFFFFFFFu);            // global_addr[31:0]
    g0.w = (unsigned)((ga >> 32) & 0x01FFFFFFu)     // global_addr[56:32]
         | (2u << 30);                              // type = 2 ("image")

    int32x8_t g1;
    g1[0] = (int)(2u << 16);                        // data_size=2 (4B); mask/pad/iterate = 0
    g1[1] = (int)((COLS & 0xFFFFu) << 16);          // tensor_dim0[15:0]
    g1[2] = (int)(((COLS >> 16) & 0xFFFFu)          // tensor_dim0[31:16]
         | (1u << 16));                             // tensor_dim1 = 1
    g1[3] = (int)((COLS & 0xFFFFu) << 16);          // tile_dim0 = 4096
    g1[4] = 1;                                      // tile_dim1 = 1, tile_dim2 = 0
    g1[5] = COLS;                                   // tensor_dim0_stride[31:0]
    g1[6] = 0;
    g1[7] = 0;

    int32x4_t gz = {0, 0, 0, 0};
#if __clang_major__ >= 23
    int32x8_t gz8 = {0, 0, 0, 0, 0, 0, 0, 0};
    __builtin_amdgcn_tensor_load_to_lds(g0, g1, gz, gz, gz8, 0);
#else
    __builtin_amdgcn_tensor_load_to_lds(g0, g1, gz, gz, 0);
#endif
    __builtin_amdgcn_s_wait_tensorcnt(0);           // wave 0 waits for DMA completion
  }
#else
  // Fallback: cooperative coalesced load of raw bits into LDS.
  {
    const float* rowp = x + (size_t)row * COLS;
    for (int j = 0; j < PER_THREAD; ++j) {
      unsigned idx = t + j * THREADS;
      s_data[idx] = __float_as_uint(rowp[idx]);
    }
  }
#endif
  __syncthreads();

  // ---- Read this thread's 16 elements ONCE; keep order-preserving keys in VGPRs.
  //      neg -> ~bits, pos -> bits | 0x80000000  (monotone: x>=y <=> u(x)>=u(y))
  unsigned key[PER_THREAD];
#pragma unroll
  for (int j = 0; j < PER_THREAD; ++j) {
    unsigned u = s_data[t + j * THREADS];
    key[j] = (u & 0x80000000u) ? ~u : (u | 0x80000000u);
  }

  // ---- 4-pass MSB-first radix select of the exact K-th largest key.
  for (int pass = 0; pass < 4; ++pass) {
    const unsigned shift  = 24u - 8u * (unsigned)pass;
    const unsigned himask = (pass == 0) ? 0u : (0xFFFFFFFFu << (shift + 8u));

    s_hist[t & 255u] = 0u;                          // 256 threads clear 256 buckets
    __syncthreads();

    const unsigned prefix = s_prefix;
    const unsigned krem   = s_krem;

#pragma unroll
    for (int j = 0; j < PER_THREAD; ++j) {
      unsigned u = key[j];
      if ((u & himask) == (prefix & himask)) {      // candidate filter for this pass
        atomicAdd(&s_hist[(u >> shift) & 0xFFu], 1u);  // ds_add_u32 (no return)
      }
    }
    __syncthreads();

    // Wave 0 alone: descending cumulative over 256 buckets via 32-lane shuffle
    // suffix-scan of 8-bucket block sums; find bucket where cum crosses krem.
    if (t < 32) {
      unsigned own = 0;
#pragma unroll
      for (int b = 0; b < 8; ++b) own += s_hist[t * 8u + b];

      unsigned sfx = own;                           // inclusive suffix over lanes
#pragma unroll
      for (int d = 1; d < 32; d <<= 1) {
        unsigned o = (unsigned)__shfl_down((int)sfx, d, 32);
        if (t + d < 32) sfx += o;
      }

      unsigned cum = sfx - own;                     // count of keys in digits > this block
      int found = -1; unsigned knext = 0;
#pragma unroll
      for (int b = 7; b >= 0; --b) {                // walk block's buckets high -> low
        unsigned before = cum;
        cum += s_hist[t * 8u + (unsigned)b];
        if (before < krem && cum >= krem) {         // unique global crossing point
          found = (int)(t * 8u + (unsigned)b);
          knext = krem - before;
        }
      }
      if (found >= 0) {                             // exactly one lane wins
        s_prefix = prefix | ((unsigned)found << shift);
        s_krem   = knext;
      }
    }
    __syncthreads();
  }

  // ---- Emit 1.0/0.0 with the exact threshold.
  const unsigned T = s_prefix;

#ifdef USE_TDM
  // Reuse the staging buffer for the result row, then DMA it out with the TDM.
#pragma unroll
  for (int j = 0; j < PER_THREAD; ++j) {
    s_data[t + j * THREADS] = (key[j] >= T) ? 0x3F800000u : 0u;  // 1.0f / 0.0f bits
  }
  __syncthreads();

  if (t < 32) {
    unsigned long long ga  = (unsigned long long)(out + (size_t)row * COLS);
    unsigned lds_off       = (unsigned)(unsigned long long)(s_data);

    uint32x4_t g0;
    g0.x = 1u;                                      // count=1
    g0.y = lds_off;                                 // lds_addr (source for store)
    g0.z = (unsigned)(ga & 0xFFFFFFFFu);
    g0.w = (unsigned)((ga >> 32) & 0x01FFFFFFu) | (2u << 30);

    int32x8_t g1;
    g1[0] = (int)(2u << 16);                        // data_size = 4B
    g1[1] = (int)((COLS & 0xFFFFu) << 16);          // tensor_dim0
    g1[2] = (int)(((COLS >> 16) & 0xFFFFu) | (1u << 16));
    g1[3] = (int)((COLS & 0xFFFFu) << 16);          // tile_dim0 = 4096
    g1[4] = 1;                                      // tile_dim1 = 1
    g1[5] = COLS;                                   // tensor_dim0_stride
    g1[6] = 0;
    g1[7] = 0;

    int32x4_t gz = {0, 0, 0, 0};
#if __clang_major__ >= 23
    int32x8_t gz8 = {0, 0, 0, 0, 0, 0, 0, 0};
    __builtin_amdgcn_tensor_store_from_lds(g0, g1, gz, gz, gz8, 0);
#else
    __builtin_amdgcn_tensor_store_from_lds(g0, g1, gz, gz, 0);
#endif
    __builtin_amdgcn_s_wait_tensorcnt(0);           // keep workgroup (and LDS) alive until done
  }
#else
  float* orow = out + (size_t)row * COLS;
#pragma unroll
  for (int j = 0; j < PER_THREAD; ++j) {
    orow[t + j * THREADS] = (key[j] >= T) ? 1.0f : 0.0f;
  }
#endif
}

extern "C" void kernel_launch(void* const* d_in, const int* in_sizes, int n_in,
                              void* d_out, int out_size, void* d_ws, size_t ws_size,
                              hipStream_t stream) {
  const float* x = (const float*)d_in[0];
  float* out     = (float*)d_out;
  int rows       = in_sizes[0] / COLS;              // 16384 for the reference shapes
  KWinnersTakeAll_kernel<<<rows, THREADS, 0, stream>>>(x, out);
}